// GIN_71880572666183
// MI455X (gfx1250) — compile-verified
//
#include <hip/hip_runtime.h>
#include <stdint.h>

#define NN    8192
#define BATCH 4
#define CIN   32
#define CHID  64
#define COUT  32
#define EPS_P1 1.01f     // 1 + eps, eps = 0.01

#define KC     64        // K-chunk staged in LDS per double-buffer step
#define APITCH 68        // padded LDS row pitch in floats (68 mod 64 = 4 -> conflict-free)

#define USE_TDM 1        // 1: tensor_load_to_lds (TDM DMA); 0: per-lane async copies

typedef __attribute__((ext_vector_type(2))) float    v2f;
typedef __attribute__((ext_vector_type(8))) float    v8f;
typedef __attribute__((ext_vector_type(4))) uint32_t u32x4;
typedef __attribute__((ext_vector_type(8))) uint32_t u32x8;

#if USE_TDM
// One TDM descriptor load: 64 rows x 64 f32 tile, row stride NN, LDS pitch
// padded to 68 floats via D# pad fields (4 DWORDs every 64 DWORDs).
__device__ __forceinline__ void tdm_load_tile(uint32_t lds_byte, const float* g) {
  const uint64_t ga = (uint64_t)(uintptr_t)g;
  u32x4 g0;
  g0[0] = 1u;                                   // count=1 (valid user D#)
  g0[1] = lds_byte;                             // lds_addr (bytes)
  g0[2] = (uint32_t)ga;                         // global_addr[31:0]
  g0[3] = (uint32_t)((ga >> 32) & 0x1FFFFFFu)   // global_addr[56:32]
        | (2u << 30);                           // type=2 ("image")
  u32x8 g1;
  g1[0] = (2u << 16)                            // data_size = 4 bytes
        | (1u << 20)                            // pad_enable
        | (5u << 22)                            // pad_interval: every 64 DWORDs
        | (3u << 25);                           // pad_amount: 4 DWORDs
  g1[1] = (uint32_t)NN << 16;                   // tensor_dim0[15:0] (=8192) in [63:48]
  g1[2] = (uint32_t)NN << 16;                   // dim0 hi=0 | tensor_dim1[15:0] in [95:80]
  g1[3] = (uint32_t)KC << 16;                   // dim1 hi=0 | tile_dim0=64 in [127:112]
  g1[4] = (uint32_t)64;                         // tile_dim1=64 rows, tile_dim2=0
  g1[5] = (uint32_t)NN;                         // tensor_dim0_stride[31:0]
  g1[6] = 0u;                                   // stride0 hi | stride1 lo
  g1[7] = 0u;                                   // stride1 hi
  asm volatile("tensor_load_to_lds %0, %1" :: "s"(g0), "s"(g1) : "memory");
}
#else
__device__ __forceinline__ void async_copy_b128(uint32_t lds_byte_off,
                                                const float* gaddr) {
  asm volatile("global_load_async_to_lds_b128 %0, %1, off"
               :: "v"(lds_byte_off),
                  "v"((unsigned long long)(uintptr_t)gaddr)
               : "memory");
}
#endif

// ---------------------------------------------------------------------------
// Stage 1 (K-split): P[ks][b,i,c] = sum_{j in split ks} A[i,j] * X[b,j,c]
// Block: 256 threads (8 waves) owns rows [bx*64, bx*64+64), K range by by.
// A strip double-buffered in LDS via the Tensor Data Mover; each wave
// (batch = w>>1, col tile = (w&1)*16) runs 4 stacked 16x16 f32 WMMA accums.
// ---------------------------------------------------------------------------
__global__ __launch_bounds__(256)
void gin_aggregate(const float* __restrict__ A, const float* __restrict__ X,
                   float* __restrict__ P, int klen) {
  __shared__ float smem[2 * 64 * APITCH];   // 34.8 KB double buffer

  const int tid  = threadIdx.x;
  const int wave = tid >> 5;
  const int lane = tid & 31;
  const int half = lane >> 4;
  const int lrow = lane & 15;
  const int koff = half * 2;

  const int b     = wave >> 1;
  const int n0    = (wave & 1) * 16;
  const int row0  = blockIdx.x * 64;
  const int kbase = blockIdx.y * klen;

  const uint32_t smem_base = (uint32_t)(uintptr_t)smem;
  const float* gA = A + (size_t)row0 * NN + kbase;

#if USE_TDM
  auto issue = [&](int kchunk, int buf) {
    tdm_load_tile(smem_base + (uint32_t)(buf * 64 * APITCH) * 4u, gA + kchunk);
  };
#else
  auto issue = [&](int kchunk, int buf) {
    const float* gb = gA + kchunk;
    const uint32_t lb = smem_base + (uint32_t)(buf * 64 * APITCH) * 4u;
#pragma unroll
    for (int i = 0; i < 4; ++i) {
      const int u  = tid + i * 256;
      const int r  = u >> 4;
      const int cs = (u & 15) * 4;
      async_copy_b128(lb + (uint32_t)(r * APITCH + cs) * 4u,
                      gb + (size_t)r * NN + cs);
    }
  };
#endif

  const float* xc = X + (size_t)b * NN * CIN + n0 + lrow;

  v8f acc[4] = {};
  const int nchunk = klen / KC;

#if USE_TDM
  if (wave == 0) issue(0, 0);
#else
  issue(0, 0);
#endif

  for (int c = 0; c < nchunk; ++c) {
#if USE_TDM
    if (wave == 0) {                       // wave-uniform branch, no barriers inside
      if (c + 1 < nchunk) {
        issue((c + 1) * KC, (c + 1) & 1);
        asm volatile("s_wait_tensorcnt 1" ::: "memory");  // chunk c landed
      } else {
        asm volatile("s_wait_tensorcnt 0" ::: "memory");
      }
    }
#else
    if (c + 1 < nchunk) {
      issue((c + 1) * KC, (c + 1) & 1);
      asm volatile("s_wait_asynccnt 4" ::: "memory");
    } else {
      asm volatile("s_wait_asynccnt 0" ::: "memory");
    }
#endif
    __syncthreads();                       // publish chunk c to all waves

    const float* lbuf = smem + (c & 1) * 64 * APITCH;
    const float* l0 = lbuf + ( 0 + lrow) * APITCH + koff;
    const float* l1 = lbuf + (16 + lrow) * APITCH + koff;
    const float* l2 = lbuf + (32 + lrow) * APITCH + koff;
    const float* l3 = lbuf + (48 + lrow) * APITCH + koff;
    const float* xj = xc + (size_t)(kbase + c * KC) * CIN;

#pragma unroll 4
    for (int j = 0; j < KC; j += 4) {
      v2f fa0 = *(const v2f*)(l0 + j);
      v2f fa1 = *(const v2f*)(l1 + j);
      v2f fa2 = *(const v2f*)(l2 + j);
      v2f fa3 = *(const v2f*)(l3 + j);
      float q0 = xj[(size_t)(j + koff)     * CIN];
      float q1 = xj[(size_t)(j + koff + 1) * CIN];
      v2f fb = {q0, q1};
      acc[0] = __builtin_amdgcn_wmma_f32_16x16x4_f32(false, fa0, false, fb,
                                                     (short)0, acc[0], false, false);
      acc[1] = __builtin_amdgcn_wmma_f32_16x16x4_f32(false, fa1, false, fb,
                                                     (short)0, acc[1], false, false);
      acc[2] = __builtin_amdgcn_wmma_f32_16x16x4_f32(false, fa2, false, fb,
                                                     (short)0, acc[2], false, false);
      acc[3] = __builtin_amdgcn_wmma_f32_16x16x4_f32(false, fa3, false, fb,
                                                     (short)0, acc[3], false, false);
    }
    __syncthreads();   // everyone done reading buf[c&1] before it is re-filled
  }

  // Store raw partial (diagonal term added in stage 2 during reduction).
  float* Pp = P + (size_t)blockIdx.y * BATCH * NN * CIN + (size_t)b * NN * CIN;
#pragma unroll
  for (int s = 0; s < 4; ++s) {
    const int rb = row0 + s * 16 + half * 8;
#pragma unroll
    for (int r = 0; r < 8; ++r)
      Pp[(size_t)(rb + r) * CIN + n0 + lrow] = acc[s][r];
  }
}

// ---------------------------------------------------------------------------
// Stage 2: Y = sum_ks P[ks] + (1+eps)*X ; out = relu(Y@W1+b1)@W2+b2
// Block: 128 threads (4 waves), one 16-row tile per wave; hidden 16x64 tile
// staged through LDS to convert WMMA C-layout -> A-fragment layout.
// ---------------------------------------------------------------------------
__global__ __launch_bounds__(128)
void gin_mlp(const float* __restrict__ P, int nsplit,
             const float* __restrict__ X,
             const float* __restrict__ W1, const float* __restrict__ B1,
             const float* __restrict__ W2, const float* __restrict__ B2,
             float* __restrict__ out) {
  __shared__ float lds[4 * 16 * 68];

  const int tid  = threadIdx.x;
  const int wave = tid >> 5;
  const int lane = tid & 31;
  const int half = lane >> 4;
  const int lrow = lane & 15;
  const int koff = half * 2;

  const int tile    = blockIdx.x * 4 + wave;
  const int rowbase = tile * 16;

  const size_t PS   = (size_t)BATCH * NN * CIN;
  const size_t roff = (size_t)(rowbase + lrow) * CIN + koff;

  // GEMM 1: H(16x64) = Y_tile(16x32) @ W1(32x64), Y reduced on the fly.
  v8f h[4] = {};
#pragma unroll
  for (int k = 0; k < CIN; k += 4) {
    v2f xv = *(const v2f*)(X + roff + k);
    v2f a;
    a.x = EPS_P1 * xv.x;
    a.y = EPS_P1 * xv.y;
    for (int s = 0; s < nsplit; ++s)
      a = a + *(const v2f*)(P + (size_t)s * PS + roff + k);
#pragma unroll
    for (int t = 0; t < 4; ++t) {
      float q0 = W1[(size_t)(k + koff)     * CHID + t * 16 + lrow];
      float q1 = W1[(size_t)(k + koff + 1) * CHID + t * 16 + lrow];
      v2f fb = {q0, q1};
      h[t] = __builtin_amdgcn_wmma_f32_16x16x4_f32(false, a, false, fb,
                                                   (short)0, h[t], false, false);
    }
  }

  // bias + ReLU -> LDS row-major [16][68]
  float* hl = lds + wave * 16 * 68;
#pragma unroll
  for (int t = 0; t < 4; ++t) {
    float bias = B1[t * 16 + lrow];
#pragma unroll
    for (int r = 0; r < 8; ++r) {
      float v = h[t][r] + bias;
      v = v > 0.0f ? v : 0.0f;
      hl[(r + half * 8) * 68 + t * 16 + lrow] = v;
    }
  }
  asm volatile("s_wait_dscnt 0" ::: "memory");   // per-wave LDS turn-around

  // GEMM 2: Z(16x32) = H(16x64) @ W2(64x32)
  const float* hr = hl + lrow * 68 + koff;
  v8f z[2] = {};
#pragma unroll
  for (int k = 0; k < CHID; k += 4) {
    v2f a = *(const v2f*)(hr + k);
#pragma unroll
    for (int u = 0; u < 2; ++u) {
      float q0 = W2[(size_t)(k + koff)     * COUT + u * 16 + lrow];
      float q1 = W2[(size_t)(k + koff + 1) * COUT + u * 16 + lrow];
      v2f fb = {q0, q1};
      z[u] = __builtin_amdgcn_wmma_f32_16x16x4_f32(false, a, false, fb,
                                                   (short)0, z[u], false, false);
    }
  }

#pragma unroll
  for (int u = 0; u < 2; ++u) {
    float bias = B2[u * 16 + lrow];
#pragma unroll
    for (int r = 0; r < 8; ++r)
      out[(size_t)(rowbase + r + half * 8) * COUT + u * 16 + lrow] = z[u][r] + bias;
  }
}

extern "C" void kernel_launch(void* const* d_in, const int* in_sizes, int n_in,
                              void* d_out, int out_size, void* d_ws, size_t ws_size,
                              hipStream_t stream) {
  const float* X  = (const float*)d_in[0];  // [4,8192,32]
  const float* A  = (const float*)d_in[1];  // [8192,8192]
  const float* W1 = (const float*)d_in[2];  // [32,64]
  const float* B1 = (const float*)d_in[3];  // [64]
  const float* W2 = (const float*)d_in[4];  // [64,32]
  const float* B2 = (const float*)d_in[5];  // [32]

  // K-split chosen from ws capacity (deterministic per process/launch config).
  const size_t PS = (size_t)BATCH * NN * CIN * sizeof(float);  // 4 MB / split
  int ksplit = 1;
  if (ws_size >= 4 * PS)      ksplit = 4;
  else if (ws_size >= 2 * PS) ksplit = 2;
  const int klen = NN / ksplit;

  dim3 g1(NN / 64, ksplit);
  gin_aggregate<<<g1, 256, 0, stream>>>(A, X, (float*)d_ws, klen);
  gin_mlp<<<(BATCH * NN / 16) / 4, 128, 0, stream>>>((float*)d_ws, ksplit, X,
                                                     W1, B1, W2, B2,
                                                     (float*)d_out);
}